// Loss_23467701305783
// MI455X (gfx1250) — compile-verified
//
#include <hip/hip_runtime.h>
#include <stdint.h>

// ---------------- problem constants ----------------
#define NPIX    147456          // 384*384
#define NSAMP   32
#define ROWS    64              // 32 region rows + 32 affinity rows
#define THREADS 256
#define BLKX    18              // 18 blocks * 256 thr * 8 iters * 4 floats = 147456
#define ITERS   8
#define H1B     512             // bits[31:21] of losses in [0,1)
#define H2B     2048            // bits[20:10]
#define H3B     1024            // bits[9:0]
#define SCALE_F   1.099511627776e12f   // 2^40 fixed point
#define INV_SCALE 9.094947017729282379e-13

#ifndef USE_ASYNC_LDS
#define USE_ASYNC_LDS 1
#endif

__device__ __forceinline__ uint32_t bin1_of(uint32_t bits) {
    uint32_t b = bits >> 21;
    return b > (H1B - 1u) ? (H1B - 1u) : b;
}
__device__ __forceinline__ unsigned long long quant(float l) {
    return (unsigned long long)(l * SCALE_F);
}

#if USE_ASYNC_LDS
// gfx1250 async global->LDS copy (ASYNCcnt-tracked). GV addressing mode:
//   global_load_async_to_lds_b128  vdst(lds byte addr), vaddr(64b global), off
__device__ __forceinline__ void async_load_b128(uint32_t lds_addr, const void* gaddr) {
    asm volatile("global_load_async_to_lds_b128 %0, %1, off"
                 :: "v"(lds_addr), "v"((unsigned long long)(uintptr_t)gaddr)
                 : "memory");
}
#endif

// ---------------- workspace zero ----------------
__global__ void k_zero(uint32_t* w, int n) {
    int i = blockIdx.x * blockDim.x + threadIdx.x;
    if (i < n) w[i] = 0u;
}

// ---------------- pass 1: loss + pos stats + level-1 histogram ----------------
__global__ void k_pass1(const float* __restrict__ gtR, const float* __restrict__ gtA,
                        const float* __restrict__ prR, const float* __restrict__ prA,
                        const float* __restrict__ mk,
                        uint32_t* __restrict__ h1, uint32_t* __restrict__ nposv,
                        unsigned long long* __restrict__ posfx) {
    __shared__ uint32_t lh[H1B];
    __shared__ uint32_t lpc;
    __shared__ unsigned long long lps;
    const int tid = threadIdx.x;
    for (int i = tid; i < H1B; i += THREADS) lh[i] = 0u;
    if (tid == 0) { lpc = 0u; lps = 0ull; }
    __syncthreads();

    const int row = blockIdx.y;
    const int s   = row & 31;
    const bool aff = row >= NSAMP;
    const float4* g = (const float4*)((aff ? gtA : gtR) + (size_t)s * NPIX);
    const float4* p = (const float4*)((aff ? prA : prR) + (size_t)s * NPIX);
    const float4* m = (const float4*)(mk + (size_t)s * NPIX);

    uint32_t pc = 0u;
    unsigned long long ps = 0ull;
    const int base = blockIdx.x * (THREADS * ITERS);

#define K1_PROC(G, P, M) do { float d = (P) - (G); float l = d * d * (M);          \
        if ((G) >= 0.1f) { pc++; ps += quant(l); }                                  \
        else { atomicAdd(&lh[bin1_of(__float_as_uint(l))], 1u); } } while (0)

    for (int it = 0; it < ITERS; ++it) {
        const int f4 = base + it * THREADS + tid;
        float4 gv = g[f4], pv = p[f4], mv = m[f4];
        K1_PROC(gv.x, pv.x, mv.x);
        K1_PROC(gv.y, pv.y, mv.y);
        K1_PROC(gv.z, pv.z, mv.z);
        K1_PROC(gv.w, pv.w, mv.w);
    }
#undef K1_PROC
    if (pc) { atomicAdd(&lpc, pc); atomicAdd(&lps, ps); }
    __syncthreads();
    for (int i = tid; i < H1B; i += THREADS)
        if (lh[i]) atomicAdd(&h1[row * H1B + i], lh[i]);
    if (tid == 0 && lpc) {
        atomicAdd(&nposv[row], lpc);
        atomicAdd(&posfx[row], lps);
    }
}

// ---------------- scan 1: pick k and coarse bin t1 ----------------
__global__ void k_scan1(const uint32_t* __restrict__ h1, const uint32_t* __restrict__ nposv,
                        uint32_t* __restrict__ kk, uint32_t* __restrict__ t1v,
                        uint32_t* __restrict__ cnt) {
    const int row = blockIdx.x;
    __shared__ uint32_t lh[H1B];
    for (int i = threadIdx.x; i < H1B; i += blockDim.x) lh[i] = h1[row * H1B + i];
    __syncthreads();
    if (threadIdx.x == 0) {
        const uint32_t npos = nposv[row];
        const uint32_t nneg = (uint32_t)NPIX - npos;
        uint32_t k;
        if (npos > 0u) {
            uint32_t kt = 3u * npos; if (kt > nneg) kt = nneg; if (kt < 1u) kt = 1u;
            k = kt;
        } else k = 500u;
        kk[row] = k;
        uint32_t cum = 0u; int t = 0;
        for (int b = H1B - 1; b >= 0; --b) {
            uint32_t c = lh[b];
            if (cum + c >= k) { t = b; break; }
            cum += c;
        }
        t1v[row] = (uint32_t)t;
        cnt[row] = cum;          // elements strictly above bin t1
    }
}

// ---------------- pass 2: mid histogram inside t1 + sum above t1 ----------------
__global__ void k_refine1(const float* __restrict__ gtR, const float* __restrict__ gtA,
                          const float* __restrict__ prR, const float* __restrict__ prA,
                          const float* __restrict__ mk,
                          const uint32_t* __restrict__ t1v,
                          uint32_t* __restrict__ h2, unsigned long long* __restrict__ sumfx) {
    __shared__ uint32_t lh[H2B];
    __shared__ unsigned long long lsum;
#if USE_ASYNC_LDS
    __shared__ float4 stage[2][3][THREADS];
#endif
    const int tid = threadIdx.x;
    for (int i = tid; i < H2B; i += THREADS) lh[i] = 0u;
    if (tid == 0) lsum = 0ull;
    __syncthreads();

    const int row = blockIdx.y;
    const int s   = row & 31;
    const bool aff = row >= NSAMP;
    const float4* g = (const float4*)((aff ? gtA : gtR) + (size_t)s * NPIX);
    const float4* p = (const float4*)((aff ? prA : prR) + (size_t)s * NPIX);
    const float4* m = (const float4*)(mk + (size_t)s * NPIX);
    const uint32_t T1 = t1v[row];
    unsigned long long acc = 0ull;
    const int base = blockIdx.x * (THREADS * ITERS);

#if USE_ASYNC_LDS
    uint32_t so[2][3];
#pragma unroll
    for (int b = 0; b < 2; ++b)
#pragma unroll
        for (int a = 0; a < 3; ++a)
            so[b][a] = (uint32_t)(uintptr_t)&stage[b][a][tid];
    {
        const int f4 = base + tid;
        async_load_b128(so[0][0], (const void*)(g + f4));
        async_load_b128(so[0][1], (const void*)(p + f4));
        async_load_b128(so[0][2], (const void*)(m + f4));
    }
#endif

#define K3_PROC(G, P, M) do { if ((G) < 0.1f) { float d = (P) - (G); float l = d * d * (M); \
        uint32_t bits = __float_as_uint(l); uint32_t b1 = bin1_of(bits);                     \
        if (b1 == T1)      atomicAdd(&lh[(bits >> 10) & (H2B - 1u)], 1u);                    \
        else if (b1 > T1)  acc += quant(l); } } while (0)

    for (int it = 0; it < ITERS; ++it) {
        float4 gv, pv, mv;
#if USE_ASYNC_LDS
        const int buf = it & 1;
        if (it + 1 < ITERS) {
            const int f4n = base + (it + 1) * THREADS + tid;
            async_load_b128(so[buf ^ 1][0], (const void*)(g + f4n));
            async_load_b128(so[buf ^ 1][1], (const void*)(p + f4n));
            async_load_b128(so[buf ^ 1][2], (const void*)(m + f4n));
            asm volatile("s_wait_asynccnt 0x3" ::: "memory");  // previous 3 done
        } else {
            asm volatile("s_wait_asynccnt 0x0" ::: "memory");
        }
        gv = stage[buf][0][tid]; pv = stage[buf][1][tid]; mv = stage[buf][2][tid];
#else
        const int f4 = base + it * THREADS + tid;
        gv = g[f4]; pv = p[f4]; mv = m[f4];
#endif
        K3_PROC(gv.x, pv.x, mv.x);
        K3_PROC(gv.y, pv.y, mv.y);
        K3_PROC(gv.z, pv.z, mv.z);
        K3_PROC(gv.w, pv.w, mv.w);
    }
#undef K3_PROC
    if (acc) atomicAdd(&lsum, acc);
    __syncthreads();
    for (int i = tid; i < H2B; i += THREADS)
        if (lh[i]) atomicAdd(&h2[row * H2B + i], lh[i]);
    if (tid == 0 && lsum) atomicAdd(&sumfx[row], lsum);
}

// ---------------- scan 2 ----------------
__global__ void k_scan2(const uint32_t* __restrict__ h2, const uint32_t* __restrict__ kk,
                        uint32_t* __restrict__ t2v, uint32_t* __restrict__ cnt) {
    const int row = blockIdx.x;
    __shared__ uint32_t lh[H2B];
    for (int i = threadIdx.x; i < H2B; i += blockDim.x) lh[i] = h2[row * H2B + i];
    __syncthreads();
    if (threadIdx.x == 0) {
        const uint32_t krem = kk[row] - cnt[row];
        uint32_t cum = 0u; int t = 0;
        for (int b = H2B - 1; b >= 0; --b) {
            uint32_t c = lh[b];
            if (cum + c >= krem) { t = b; break; }
            cum += c;
        }
        t2v[row] = (uint32_t)t;
        cnt[row] += cum;
    }
}

// ---------------- pass 3: low-bit histogram inside (t1,t2) + sum above t2 ----------------
__global__ void k_refine2(const float* __restrict__ gtR, const float* __restrict__ gtA,
                          const float* __restrict__ prR, const float* __restrict__ prA,
                          const float* __restrict__ mk,
                          const uint32_t* __restrict__ t1v, const uint32_t* __restrict__ t2v,
                          uint32_t* __restrict__ h3, unsigned long long* __restrict__ sumfx) {
    __shared__ uint32_t lh[H3B];
    __shared__ unsigned long long lsum;
#if USE_ASYNC_LDS
    __shared__ float4 stage[2][3][THREADS];
#endif
    const int tid = threadIdx.x;
    for (int i = tid; i < H3B; i += THREADS) lh[i] = 0u;
    if (tid == 0) lsum = 0ull;
    __syncthreads();

    const int row = blockIdx.y;
    const int s   = row & 31;
    const bool aff = row >= NSAMP;
    const float4* g = (const float4*)((aff ? gtA : gtR) + (size_t)s * NPIX);
    const float4* p = (const float4*)((aff ? prA : prR) + (size_t)s * NPIX);
    const float4* m = (const float4*)(mk + (size_t)s * NPIX);
    const uint32_t T1 = t1v[row];
    const uint32_t T2 = t2v[row];
    unsigned long long acc = 0ull;
    const int base = blockIdx.x * (THREADS * ITERS);

#if USE_ASYNC_LDS
    uint32_t so[2][3];
#pragma unroll
    for (int b = 0; b < 2; ++b)
#pragma unroll
        for (int a = 0; a < 3; ++a)
            so[b][a] = (uint32_t)(uintptr_t)&stage[b][a][tid];
    {
        const int f4 = base + tid;
        async_load_b128(so[0][0], (const void*)(g + f4));
        async_load_b128(so[0][1], (const void*)(p + f4));
        async_load_b128(so[0][2], (const void*)(m + f4));
    }
#endif

#define K5_PROC(G, P, M) do { if ((G) < 0.1f) { float d = (P) - (G); float l = d * d * (M); \
        uint32_t bits = __float_as_uint(l); uint32_t b1 = bin1_of(bits);                     \
        if (b1 == T1) { uint32_t mid = (bits >> 10) & (H2B - 1u);                            \
            if (mid == T2)      atomicAdd(&lh[bits & (H3B - 1u)], 1u);                       \
            else if (mid > T2)  acc += quant(l); } } } while (0)

    for (int it = 0; it < ITERS; ++it) {
        float4 gv, pv, mv;
#if USE_ASYNC_LDS
        const int buf = it & 1;
        if (it + 1 < ITERS) {
            const int f4n = base + (it + 1) * THREADS + tid;
            async_load_b128(so[buf ^ 1][0], (const void*)(g + f4n));
            async_load_b128(so[buf ^ 1][1], (const void*)(p + f4n));
            async_load_b128(so[buf ^ 1][2], (const void*)(m + f4n));
            asm volatile("s_wait_asynccnt 0x3" ::: "memory");
        } else {
            asm volatile("s_wait_asynccnt 0x0" ::: "memory");
        }
        gv = stage[buf][0][tid]; pv = stage[buf][1][tid]; mv = stage[buf][2][tid];
#else
        const int f4 = base + it * THREADS + tid;
        gv = g[f4]; pv = p[f4]; mv = m[f4];
#endif
        K5_PROC(gv.x, pv.x, mv.x);
        K5_PROC(gv.y, pv.y, mv.y);
        K5_PROC(gv.z, pv.z, mv.z);
        K5_PROC(gv.w, pv.w, mv.w);
    }
#undef K5_PROC
    if (acc) atomicAdd(&lsum, acc);
    __syncthreads();
    for (int i = tid; i < H3B; i += THREADS)
        if (lh[i]) atomicAdd(&h3[row * H3B + i], lh[i]);
    if (tid == 0 && lsum) atomicAdd(&sumfx[row], lsum);
}

// ---------------- scan 3: exact kth value + tail sum ----------------
__global__ void k_scan3(const uint32_t* __restrict__ h3, const uint32_t* __restrict__ kk,
                        const uint32_t* __restrict__ t1v, const uint32_t* __restrict__ t2v,
                        uint32_t* __restrict__ cnt, double* __restrict__ sumC,
                        float* __restrict__ kthv) {
    const int row = blockIdx.x;
    __shared__ uint32_t lh[H3B];
    for (int i = threadIdx.x; i < H3B; i += blockDim.x) lh[i] = h3[row * H3B + i];
    __syncthreads();
    if (threadIdx.x == 0) {
        const uint32_t hi = (t1v[row] << 21) | (t2v[row] << 10);
        const uint32_t krem = kk[row] - cnt[row];
        uint32_t cum = 0u; int t = 0; double sc = 0.0;
        for (int b = H3B - 1; b >= 0; --b) {
            uint32_t c = lh[b];
            if (cum + c >= krem) { t = b; break; }
            cum += c;
            // all elements in this bin share one exact fp32 value
            sc += (double)c * (double)__uint_as_float(hi | (uint32_t)b);
        }
        cnt[row] += cum;
        sumC[row] = sc;
        kthv[row] = __uint_as_float(hi | (uint32_t)t);
    }
}

// ---------------- finalize ----------------
__global__ void k_final(const uint32_t* __restrict__ nposv, const uint32_t* __restrict__ kk,
                        const uint32_t* __restrict__ cnt,
                        const unsigned long long* __restrict__ posfx,
                        const unsigned long long* __restrict__ sumfx,
                        const double* __restrict__ sumC, const float* __restrict__ kthv,
                        float* __restrict__ out) {
    __shared__ double acc[ROWS];
    const int r = threadIdx.x;
    const uint32_t npos = nposv[r];
    const uint32_t k    = kk[r];
    const uint32_t krem = k - cnt[r];                 // ties at kth value
    const double sum_gt = (double)sumfx[r] * INV_SCALE + sumC[r];
    const double topk   = sum_gt + (double)krem * (double)kthv[r];
    const double pt     = npos ? ((double)posfx[r] * INV_SCALE) / (double)npos : 0.0;
    acc[r] = pt + topk / (double)k;
    __syncthreads();
    if (r == 0) {
        double t = 0.0;
        for (int i = 0; i < ROWS; ++i) t += acc[i];
        out[0] = (float)(t / (double)NSAMP);
    }
}

// ---------------- launch ----------------
extern "C" void kernel_launch(void* const* d_in, const int* in_sizes, int n_in,
                              void* d_out, int out_size, void* d_ws, size_t ws_size,
                              hipStream_t stream) {
    (void)in_sizes; (void)n_in; (void)out_size; (void)ws_size;
    const float* gtR = (const float*)d_in[0];
    const float* gtA = (const float*)d_in[1];
    const float* prR = (const float*)d_in[2];
    const float* prA = (const float*)d_in[3];
    const float* mk  = (const float*)d_in[4];

    char* w = (char*)d_ws;
    uint32_t* h1 = (uint32_t*)(w);                                   // 131072 B
    uint32_t* h2 = (uint32_t*)(w + 131072);                          // 524288 B
    uint32_t* h3 = (uint32_t*)(w + 655360);                          // 262144 B
    char* st = w + 917504;                                           // stats block
    uint32_t* npos = (uint32_t*)(st + 0);
    uint32_t* kk   = (uint32_t*)(st + 256);
    uint32_t* t1   = (uint32_t*)(st + 512);
    uint32_t* t2   = (uint32_t*)(st + 768);
    uint32_t* cnt  = (uint32_t*)(st + 1024);
    unsigned long long* posfx = (unsigned long long*)(st + 1280);
    unsigned long long* sumfx = (unsigned long long*)(st + 1792);
    double* sumC = (double*)(st + 2304);
    float*  kth  = (float*)(st + 2816);

    const int zwords = (917504 + 3072) / 4;
    k_zero<<<(zwords + 255) / 256, 256, 0, stream>>>((uint32_t*)w, zwords);

    dim3 grid(BLKX, ROWS);
    k_pass1  <<<grid, THREADS, 0, stream>>>(gtR, gtA, prR, prA, mk, h1, npos, posfx);
    k_scan1  <<<ROWS, 256, 0, stream>>>(h1, npos, kk, t1, cnt);
    k_refine1<<<grid, THREADS, 0, stream>>>(gtR, gtA, prR, prA, mk, t1, h2, sumfx);
    k_scan2  <<<ROWS, 256, 0, stream>>>(h2, kk, t2, cnt);
    k_refine2<<<grid, THREADS, 0, stream>>>(gtR, gtA, prR, prA, mk, t1, t2, h3, sumfx);
    k_scan3  <<<ROWS, 256, 0, stream>>>(h3, kk, t1, t2, cnt, sumC, kth);
    k_final  <<<1, ROWS, 0, stream>>>(npos, kk, cnt, posfx, sumfx, sumC, kth, (float*)d_out);
}